// SolveGradientsLST_88510686036692
// MI455X (gfx1250) — compile-verified
//
#include <hip/hip_runtime.h>
#include <hip/hip_bf16.h>
#include <stdint.h>

// ---------------- tunables ----------------
#define TILE    2048      // edges staged per pipeline stage
#define THREADS 256       // 8 waves of 32 on CDNA5
#define GRADLIM 30000.0f
#define EPSV    1e-8f

// ---------------- CDNA5 async global->LDS helpers ----------------
// global_load_async_to_lds_b128: VDST = per-lane LDS byte address (wave-relative;
// truncating the flat shared pointer yields the LDS offset per CDNA5 ISA 10.2),
// VADDR = per-lane 64-bit global address. Tracked with ASYNCcnt.
__device__ __forceinline__ void async_load_b128(const int* lds_chunk, const int* gsrc) {
    unsigned           off = (unsigned)(uintptr_t)lds_chunk;          // LDS byte offset
    unsigned long long ga  = (unsigned long long)(uintptr_t)gsrc;     // global address
    asm volatile("global_load_async_to_lds_b128 %0, %1, off"
                 :
                 : "v"(off), "v"(ga)
                 : "memory");
}

__device__ __forceinline__ void wait_asynccnt0() {
    asm volatile("s_wait_asynccnt 0x0" ::: "memory");
}

// Read two staged indices straight off the DS pipe. LLVM cannot see the async
// writes into shared memory, so the reads must be explicit ds_load asm (volatile
// C++ reads were lowered as FLAT loads in round 3). "=&v" earlyclobber: the data
// return of load #1 may land before load #2 issues, so outputs must not alias
// the address inputs.
__device__ __forceinline__ void lds_read_pair(const int* rowp, const int* colp,
                                              int& r, int& c) {
    unsigned orow = (unsigned)(uintptr_t)rowp;
    unsigned ocol = (unsigned)(uintptr_t)colp;
    int vr, vc;
    asm volatile("ds_load_b32 %0, %2\n\t"
                 "ds_load_b32 %1, %3\n\t"
                 "s_wait_dscnt 0x0"
                 : "=&v"(vr), "=&v"(vc)
                 : "v"(orow), "v"(ocol)
                 : "memory");
    r = vr;
    c = vc;
}

// ---------------- kernel 1: zero the accumulator workspace (b128 stores) -----
__global__ __launch_bounds__(THREADS) void zero_ws(float4* __restrict__ ws4, int n4) {
    int i = blockIdx.x * blockDim.x + threadIdx.x;
    const int stride = gridDim.x * blockDim.x;
    const float4 z = make_float4(0.f, 0.f, 0.f, 0.f);
    for (; i < n4; i += stride) ws4[i] = z;
}

// ---------------- kernel 2: edge scatter with async double-buffered staging ----
// ws layout (floats, SoA): Mxx[N] | Mxy[N] | Myy[N] | Vx0..Vx3[N each] | Vy0..Vy3[N each]
__global__ __launch_bounds__(THREADS) void edge_scatter(
        const float2* __restrict__ pos,
        const int*    __restrict__ ei,      // rows at [0,E), cols at [E,2E)
        const float4* __restrict__ field,
        float*        __restrict__ ws,
        int N, int E, int nstages)
{
    __shared__ int sidx[2][2 * TILE];   // [buffer][rows(0..TILE-1) | cols(TILE..2*TILE-1)]
    const int t = threadIdx.x;

    float* Mxx = ws;
    float* Mxy = ws + (size_t)N;
    float* Myy = ws + 2 * (size_t)N;
    float* Vx  = ws + 3 * (size_t)N;    // Vx[f*N + n]
    float* Vy  = ws + 7 * (size_t)N;    // Vy[f*N + n]

    auto issue_stage = [&](int s, int b) {
        const int base = s * TILE;
        // TILE*4 bytes per index array; THREADS lanes x 16B per issue -> 2 issues/array
        #pragma unroll
        for (int j = 0; j < (TILE * 4) / (THREADS * 16); ++j) {
            const int elt = (j * THREADS + t) * 4;            // 4 ints per 16B chunk
            async_load_b128(&sidx[b][elt],        ei + base + elt);
            async_load_b128(&sidx[b][TILE + elt], ei + E + base + elt);
        }
    };

    int s = blockIdx.x;
    if (s >= nstages) return;

    if (s * TILE + TILE <= E) issue_stage(s, 0);   // prime the pipeline (full stages only)

    int i = 0;
    for (; s < nstages; s += gridDim.x, ++i) {
        const int b = i & 1;

        wait_asynccnt0();      // my async issues for this stage are done
        __syncthreads();       // everyone's are -> buffer b valid; buffer b^1 free

        const int snext = s + (int)gridDim.x;
        if (snext < nstages && snext * TILE + TILE <= E)
            issue_stage(snext, b ^ 1);             // overlap next stage's DMA with compute

        const int  base = s * TILE;
        const int  cnt  = min(TILE, E - base);
        const bool full = (cnt == TILE);

        for (int k = t; k < cnt; k += THREADS) {
            int r, c;
            if (full) lds_read_pair(&sidx[b][k], &sidx[b][TILE + k], r, c);
            else      { r = ei[base + k]; c = ei[E + base + k]; }   // ragged tail stage

            const float2 pr = pos[r], pc = pos[c];
            const float dx = pc.x - pr.x;
            const float dy = pc.y - pr.y;
            atomicAdd(&Mxx[r], dx * dx);
            atomicAdd(&Mxy[r], dx * dy);
            atomicAdd(&Myy[r], dy * dy);

            const float4 fr = field[r], fc = field[c];
            const float d0 = fc.x - fr.x;
            const float d1 = fc.y - fr.y;
            const float d2 = fc.z - fr.z;
            const float d3 = fc.w - fr.w;
            atomicAdd(&Vx[0 * N + r], dx * d0);
            atomicAdd(&Vx[1 * N + r], dx * d1);
            atomicAdd(&Vx[2 * N + r], dx * d2);
            atomicAdd(&Vx[3 * N + r], dx * d3);
            atomicAdd(&Vy[0 * N + r], dy * d0);
            atomicAdd(&Vy[1 * N + r], dy * d1);
            atomicAdd(&Vy[2 * N + r], dy * d2);
            atomicAdd(&Vy[3 * N + r], dy * d3);
        }
        // no trailing barrier needed: top-of-loop barrier orders buffer reuse
    }
}

// ---------------- kernel 3: per-node 2x2 solve + clip + transpose-store ------
__global__ __launch_bounds__(THREADS) void solve_nodes(
        const float* __restrict__ ws, float* __restrict__ out, int N)
{
    const int n = blockIdx.x * blockDim.x + threadIdx.x;
    if (n >= N) return;

    const float mxx = ws[n] + EPSV;
    const float mxy = ws[(size_t)N + n];
    const float myy = ws[2 * (size_t)N + n] + EPSV;
    const float det = mxx * myy - mxy * mxy;
    const float inv = 1.0f / det;

    #pragma unroll
    for (int f = 0; f < 4; ++f) {
        const float vx = ws[(size_t)(3 + f) * N + n];
        const float vy = ws[(size_t)(7 + f) * N + n];
        float gx = (myy * vx - mxy * vy) * inv;
        float gy = (mxx * vy - mxy * vx) * inv;
        gx = fminf(fmaxf(gx, -GRADLIM), GRADLIM);
        gy = fminf(fmaxf(gy, -GRADLIM), GRADLIM);
        // out is [F, N, 2] flattened
        ((float2*)out)[(size_t)f * N + n] = make_float2(gx, gy);
    }
}

// ---------------- launcher ----------------
extern "C" void kernel_launch(void* const* d_in, const int* in_sizes, int n_in,
                              void* d_out, int out_size, void* d_ws, size_t ws_size,
                              hipStream_t stream) {
    const float* pos   = (const float*)d_in[0];   // [N,2] f32
    const int*   ei    = (const int*)  d_in[1];   // [2,E] i32
    const float* field = (const float*)d_in[2];   // [N,4] f32

    const int N = in_sizes[0] / 2;
    const int E = in_sizes[1] / 2;

    float* ws = (float*)d_ws;                     // needs 11*N floats (~22 MB)
    const long long wsN = 11LL * (long long)N;
    const int n4 = (int)((wsN + 3) / 4);          // float4 chunks (ws_size >> this)

    int zgrid = (n4 + THREADS - 1) / THREADS;
    if (zgrid > 4096) zgrid = 4096;
    zero_ws<<<zgrid, THREADS, 0, stream>>>((float4*)ws, n4);

    const int nstages = (E + TILE - 1) / TILE;
    const int grid = (nstages < 1024) ? nstages : 1024;   // >=3 stages/block -> pipeline overlap
    edge_scatter<<<grid, THREADS, 0, stream>>>(
        (const float2*)pos, ei, (const float4*)field, ws, N, E, nstages);

    const int grid2 = (N + THREADS - 1) / THREADS;
    solve_nodes<<<grid2, THREADS, 0, stream>>>(ws, (float*)d_out, N);
}